// drrpw_net_16612933501545
// MI455X (gfx1250) — compile-verified
//
#include <hip/hip_runtime.h>
#include <math.h>

typedef __attribute__((ext_vector_type(2))) float v2f;
typedef __attribute__((ext_vector_type(8))) float v8f;

constexpr int   N      = 100;
constexpr int   PAD    = 105;   // odd-ish stride: row-stride mod 64 = 41 (odd) -> conflict-free strided row access
constexpr int   NTILE  = 7;     // ceil(100/16)
constexpr int   QROWS  = 112;   // zero-padded rows for WMMA tiles
constexpr float TOLF   = 1e-3f;
constexpr int   MAX_IT = 100;

// LDS layout (floats):
//   [    0, 11760) : Q staging (112 x 105)  -- reused as H (100 x 105) after Sigma is built
//   [11760, 22260) : Sigma (100 x 105)
//   [22260, 22360) : phi
//   [22360, 22460) : sp = Sigma @ phi
//   [22460, 22560) : g  (gradient)
//   [22560, 22660) : x  (solve rhs/solution)
//   [22660, 22672) : scalars: 0=d 1=1+d 2=risk 3=norm 4=gnorm 5=nanflag
__global__ __launch_bounds__(256) void drrpw_newton_kernel(
    const float* __restrict__ delta, const float* __restrict__ Qg,
    const float* __restrict__ prev_phi, float* __restrict__ out)
{
  __shared__ float smem[22672];
  float* Qs  = smem;              // aliases Hm (Q dead after Sigma)
  float* Hm  = smem;
  float* Sig = smem + 11760;
  float* phi = smem + 22260;
  float* sp  = smem + 22360;
  float* g   = smem + 22460;
  float* xv  = smem + 22560;
  float* sc  = smem + 22660;

  const int b    = blockIdx.x;
  const int tid  = threadIdx.x;
  const int wave = tid >> 5;
  const int lane = tid & 31;

  // ---- stage Q[b] into LDS, zero-pad rows 100..111 for WMMA tiles ----
  const float* Qb = Qg + (size_t)b * (N * N);
  for (int idx = tid; idx < QROWS * N; idx += 256) {
    int r = idx / N, c = idx - r * N;
    Qs[r * PAD + c] = (r < N) ? Qb[r * N + c] : 0.0f;
  }
  if (tid < N) phi[tid] = prev_phi[(size_t)b * N + tid];
  if (tid == 0) {
    float d = delta[b];
    sc[0] = d;
    sc[1] = 1.0f + d;
    sc[4] = sqrtf((float)(N + 1));  // torch init: ||ones(n+1)||
    sc[5] = 0.0f;                   // nan flag
  }
  __syncthreads();

  // ---- Sigma = Q @ Q^T via V_WMMA_F32_16X16X4_F32 (49 tiles over 8 waves) ----
  for (int t = wave; t < NTILE * NTILE; t += 8) {   // wave-uniform bounds: EXEC all-1s at WMMA
    int it = t / NTILE, jt = t - it * NTILE;
    v8f acc = {};
#if __has_builtin(__builtin_amdgcn_wmma_f32_16x16x4_f32)
    {
      // 32-bit A 16x4 layout: lanes 0-15 hold K=0/1 in v[0]/v[1], lanes 16-31 hold K=2/3.
      // B 4x16 mirrors it; since B = (Q tile jt)^T, per-lane addressing equals A's with jt.
      int m  = lane & 15;
      int kh = (lane >> 4) << 1;                    // 0 or 2
      const float* Ar = Qs + (it * 16 + m) * PAD;
      const float* Br = Qs + (jt * 16 + m) * PAD;
      for (int k = 0; k < N; k += 4) {
        v2f a, bb;
        a.x  = Ar[k + kh];  a.y  = Ar[k + kh + 1];
        bb.x = Br[k + kh];  bb.y = Br[k + kh + 1];
        acc = __builtin_amdgcn_wmma_f32_16x16x4_f32(
            false, a, false, bb, (short)0, acc, false, false);
      }
    }
#else
    {
      // scalar fallback producing the same C/D lane layout
      int nn2 = jt * 16 + (lane & 15);
      for (int r = 0; r < 8; ++r) {
        int mm = it * 16 + ((lane >> 4) << 3) + r;
        const float* ar = Qs + mm * PAD;
        const float* br = Qs + nn2 * PAD;
        float s = 0.f;
        for (int k = 0; k < N; ++k) s += ar[k] * br[k];
        acc[r] = s;
      }
    }
#endif
    // C/D layout: VGPR r -> M = r + 8*(lane>=16), N = lane%16
    int nn = jt * 16 + (lane & 15);
    int mb = it * 16 + ((lane >> 4) << 3);
    if (nn < N) {
      for (int r = 0; r < 8; ++r) {
        int mm = mb + r;
        if (mm < N) Sig[mm * PAD + nn] = acc[r];
      }
    }
  }
  __syncthreads();

  // ---- Newton loop, fully LDS-resident ----
  for (int iter = 0; iter < MAX_IT; ++iter) {
    if (sc[4] <= TOLF || sc[5] != 0.0f) break;      // uniform across block

    // sp = Sigma @ phi : wave-per-row, wave32 shuffle reduction
    for (int i = wave; i < N; i += 8) {
      const float* row = Sig + i * PAD;
      float s = 0.f;
      for (int j = lane; j < N; j += 32) s += row[j] * phi[j];
      for (int o = 16; o > 0; o >>= 1) s += __shfl_down(s, o, 32);
      if (lane == 0) sp[i] = s;
    }
    __syncthreads();

    // risk = phi^T sp ; norm = ||phi||  (wave 0)
    if (wave == 0) {
      float r = 0.f, q = 0.f;
      for (int i = lane; i < N; i += 32) { float p = phi[i]; r += p * sp[i]; q += p * p; }
      for (int o = 16; o > 0; o >>= 1) { r += __shfl_down(r, o, 32); q += __shfl_down(q, o, 32); }
      if (lane == 0) { sc[2] = r; sc[3] = sqrtf(q); }
    }
    __syncthreads();

    const float d = sc[0], k1 = sc[1], risk = sc[2], nrm = sc[3];

    // gradient
    if (tid < N) {
      float p = phi[tid], s = sp[tid];
      g[tid] = 2.0f * (s + d * risk / nrm * p + d * nrm / risk * s + d * d * p) - k1 / p;
    }
    __syncthreads();

    // gnorm for next iteration's check (wave 0; consumed after loop-end barrier)
    if (wave == 0) {
      float q = 0.f;
      for (int i = lane; i < N; i += 32) { float gv = g[i]; q += gv * gv; }
      for (int o = 16; o > 0; o >>= 1) q += __shfl_down(q, o, 32);
      if (lane == 0) sc[4] = sqrtf(q);
    }

    // Hessian: H = 2S + 2d(Am+Bm) + diag(k1/phi^2 + 2d^2); symmetric, effectively SPD
    {
      const float sr  = sqrtf(risk);
      const float c1  = 1.0f / (nrm * sr);
      const float c2  = nrm / (risk * sr);
      const float c3  = nrm / sr;
      const float c4  = sr / nrm;
      const float c5  = sr / (nrm * nrm * nrm);
      const float twd = 2.0f * d;
      const float dd2 = 2.0f * d * d;
      for (int f = tid; f < N * N; f += 256) {
        int i = f / N, j = f - i * N;
        float pi = phi[i], pj = phi[j], si = sp[i], sj = sp[j];
        float sg = Sig[i * PAD + j];
        float am = pi * sj * c1 - c5 * pi * pj;
        float bm = si * pj * c1 - c2 * si * sj + c3 * sg;
        float h  = 2.0f * sg + twd * (am + bm);
        if (i == j) h += twd * c4 + k1 / (pi * pi) + dd2;
        Hm[i * PAD + j] = h;
      }
    }
    if (tid < N) xv[tid] = g[tid];
    __syncthreads();

    // forward elimination (no pivoting; SPD): wave-per-row, lanes across columns (+rhs as col N)
    for (int kk = 0; kk < N; ++kk) {
      const float pinv = 1.0f / Hm[kk * PAD + kk];    // LDS broadcast
      const float* prow = Hm + kk * PAD;
      for (int i = kk + 1 + wave; i < N; i += 8) {
        float factor = Hm[i * PAD + kk] * pinv;
        float* irow = Hm + i * PAD;
        for (int j = kk + 1 + lane; j <= N; j += 32) {
          if (j < N) irow[j] -= factor * prow[j];
          else       xv[i]  -= factor * xv[kk];       // rhs column
        }
      }
      __syncthreads();
    }

    // back substitution: single wave, lockstep; volatile for cross-lane LDS carry
    if (wave == 0) {
      volatile float* vx = xv;
      for (int i = N - 1; i >= 0; --i) {
        const float* row = Hm + i * PAD;
        float s = 0.f;
        for (int j = i + 1 + lane; j < N; j += 32) s += row[j] * vx[j];
        for (int o = 16; o > 0; o >>= 1) s += __shfl_down(s, o, 32);
        if (lane == 0) vx[i] = (vx[i] - s) / row[i];
        __builtin_amdgcn_wave_barrier();
      }
    }
    __syncthreads();

    // phi -= H^{-1} g
    if (tid < N) phi[tid] -= xv[tid];
    __syncthreads();

    // NaN flag on updated phi (wave 0)
    if (wave == 0) {
      float nv = 0.f;
      for (int i = lane; i < N; i += 32) { float p = phi[i]; nv += (p != p) ? 1.f : 0.f; }
      for (int o = 16; o > 0; o >>= 1) nv += __shfl_down(nv, o, 32);
      if (lane == 0) sc[5] = nv;
    }
    __syncthreads();
  }

  if (tid < N) out[(size_t)b * N + tid] = phi[tid];
}

extern "C" void kernel_launch(void* const* d_in, const int* in_sizes, int n_in,
                              void* d_out, int out_size, void* d_ws, size_t ws_size,
                              hipStream_t stream) {
  (void)n_in; (void)out_size; (void)d_ws; (void)ws_size;
  const float* delta = (const float*)d_in[0];   // [B,1]
  const float* Q     = (const float*)d_in[1];   // [B,100,100]
  const float* pphi  = (const float*)d_in[2];   // [B,100,1]
  float* out         = (float*)d_out;           // [B,100,1]
  const int B = in_sizes[0];                    // delta flat count == B
  drrpw_newton_kernel<<<dim3(B), dim3(256), 0, stream>>>(delta, Q, pphi, out);
}